// LRU_45732811768496
// MI455X (gfx1250) — compile-verified
//
#include <hip/hip_runtime.h>
#include <stdint.h>

#define B_   8
#define L_   8192
#define H_   128
#define N_   256
#define O_   128
#define T_CH 256
#define NCH  (L_ / T_CH)   // 32 chunks per batch
#define SUB  64            // time rows per GEMM/scan sub-step

typedef __attribute__((ext_vector_type(2))) float v2f;
typedef __attribute__((ext_vector_type(8))) float v8f;
typedef unsigned int u32;
typedef __attribute__((ext_vector_type(4))) u32 v4u;
typedef __attribute__((ext_vector_type(8))) int v8i;
typedef __attribute__((ext_vector_type(4))) int v4i;

#if __has_builtin(__builtin_amdgcn_tensor_load_to_lds)
#define HAS_TDM 1
#endif

__device__ __forceinline__ void wait_tensorcnt0() {
#if __has_builtin(__builtin_amdgcn_s_wait_tensorcnt)
  __builtin_amdgcn_s_wait_tensorcnt(0);
#else
  asm volatile("s_wait_tensorcnt 0" ::: "memory");
#endif
}

#ifdef HAS_TDM
// 1-D contiguous TDM copy: nElems elements of (1<<dataSizeCode) bytes (<65536).
// D# layout per CDNA5 ISA (groups 2/3 zero => 2D-max).
// 6-arg builtin variant (clang-23 / therock-10.0 headers).
__device__ __forceinline__ void tdm_copy_to_lds(void* ldsDst, const void* gSrc,
                                                u32 nElems, u32 dataSizeCode) {
  const u32 ldsOff = (u32)(uintptr_t)ldsDst;  // low 32 bits of flat addr = LDS byte offset
  const unsigned long long ga = (unsigned long long)(uintptr_t)gSrc;
  v4u g0;
  g0[0] = 1u;                                            // count=1, user descriptor
  g0[1] = ldsOff;                                        // lds_addr
  g0[2] = (u32)ga;                                       // global_addr[31:0]
  g0[3] = (u32)((ga >> 32) & 0x01FFFFFFu) | (2u << 30);  // global_addr[56:32] | type=2
  v8i g1;
  g1[0] = (int)(dataSizeCode << 16);                     // wg_mask=0, data_size
  g1[1] = (int)((nElems & 0xFFFFu) << 16);               // tensor_dim0[15:0] @ bits63:48
  g1[2] = (int)(((nElems >> 16) & 0xFFFFu) | (1u << 16));// tensor_dim0[31:16], tensor_dim1=1
  g1[3] = (int)((nElems & 0xFFFFu) << 16);               // tile_dim0 @ bits127:112
  g1[4] = 0;                                             // tile_dim1=0, tile_dim2=0 (1-D tile)
  g1[5] = (int)nElems;                                   // tensor_dim0_stride[31:0]
  g1[6] = 0;
  g1[7] = 0;
  const v4i zero4 = {0, 0, 0, 0};
  const v8i zero8 = {0, 0, 0, 0, 0, 0, 0, 0};
  __builtin_amdgcn_tensor_load_to_lds(g0, g1, zero4, zero4, zero8, 0);
}
#endif

// ---------------------------------------------------------------------------
// k0: prep — scale Bp by exp(gamma_log); build lambda power table lam^k, k=0..T
// ---------------------------------------------------------------------------
__global__ __launch_bounds__(256) void lru_prep(
    const float* __restrict__ nu_log, const float* __restrict__ theta_log,
    const float* __restrict__ gamma_log,
    const float* __restrict__ Bp_re, const float* __restrict__ Bp_im,
    float* __restrict__ Bsc_re, float* __restrict__ Bsc_im,
    float2* __restrict__ lampow) {
  const int n = threadIdx.x;  // 256 threads == N_
  const float g = expf(gamma_log[n]);
  for (int h = 0; h < H_; ++h) {
    Bsc_re[n * H_ + h] = g * Bp_re[n * H_ + h];
    Bsc_im[n * H_ + h] = g * Bp_im[n * H_ + h];
  }
  const float rad = expf(-expf(nu_log[n]));
  const float ph  = expf(theta_log[n]);
  const float2 lam = make_float2(rad * cosf(ph), rad * sinf(ph));
  float2 p = make_float2(1.f, 0.f);
  lampow[n] = p;  // k = 0
  for (int k = 1; k <= T_CH; ++k) {
    p = make_float2(p.x * lam.x - p.y * lam.y, p.x * lam.y + p.y * lam.x);
    lampow[(size_t)k * N_ + n] = p;
  }
}

// ---------------------------------------------------------------------------
// k1: Bu GEMM (f32 WMMA 16x16x4) + chunk-local scan.  u tile staged via TDM.
// Grid: B_*NCH blocks of 256 threads (8 waves). Writes LOCAL states into the
// final states buffer at [b][t+1][n] (fixed up later by k3a).
// ---------------------------------------------------------------------------
__global__ __launch_bounds__(256) void lru_gemm_scan(
    const float* __restrict__ u, const float* __restrict__ Bsc_re,
    const float* __restrict__ Bsc_im, const float2* __restrict__ lampow,
    float2* __restrict__ states) {
  __shared__ float  uLds[SUB * H_];    // 32 KB
  __shared__ float2 buLds[SUB * N_];   // 128 KB
  const int tid  = threadIdx.x;
  const int b    = blockIdx.x / NCH;
  const int c    = blockIdx.x % NCH;
  const int lane = tid & 31;
  const int w    = tid >> 5;
  const int hi   = (lane >> 4) & 1;    // lanes 16-31 hold K+2 / M+8 halves
  const int l15  = lane & 15;
  const int tbase = c * T_CH;

  // warm L2-resident B matrices (256 KB total, stride 512 B covers them)
  __builtin_prefetch(Bsc_re + (size_t)tid * 128, 0, 0);
  __builtin_prefetch(Bsc_im + (size_t)tid * 128, 0, 0);

  const float2 lam = lampow[N_ + tid]; // lam^1 for channel 'tid'
  float2 x = make_float2(0.f, 0.f);    // chunk-local running state

  for (int s = 0; s < T_CH / SUB; ++s) {
    __syncthreads();  // previous sub-step's LDS consumers done
    // ---- stage u rows [tbase + s*SUB, +SUB): contiguous 32 KB ----
    const float* usrc = u + ((size_t)b * L_ + tbase + s * SUB) * H_;
#ifdef HAS_TDM
    if (w == 0) {                      // one wave issues the DMA
      tdm_copy_to_lds(uLds, usrc, SUB * H_, 2u /*4B elems*/);
      wait_tensorcnt0();
    }
#else
    {
      const float4* usrc4 = (const float4*)usrc;
      float4* uLds4 = (float4*)uLds;
      for (int i = tid; i < SUB * H_ / 4; i += 256) uLds4[i] = usrc4[i];
    }
#endif
    __syncthreads();

    // ---- WMMA: D(16x16) tiles, A=u rows (M=time,K=H), B=Bsc (K=H,N=chan) ----
    const int m0    = (w & 3) * 16;
    const int nbase = (w >> 2) * 128;
    const float* arow = &uLds[(m0 + l15) * H_ + 2 * hi];
    for (int j = 0; j < 8; ++j) {
      const int n0 = nbase + j * 16;
      v8f accR = {0.f, 0.f, 0.f, 0.f, 0.f, 0.f, 0.f, 0.f};
      v8f accI = {0.f, 0.f, 0.f, 0.f, 0.f, 0.f, 0.f, 0.f};
      const float* brrow = &Bsc_re[(size_t)(n0 + l15) * H_ + 2 * hi];
      const float* birow = &Bsc_im[(size_t)(n0 + l15) * H_ + 2 * hi];
      for (int k0 = 0; k0 < H_; k0 += 4) {
        v2f a  = *(const v2f*)(arow + k0);
        v2f br = *(const v2f*)(brrow + k0);
        v2f bi = *(const v2f*)(birow + k0);
        accR = __builtin_amdgcn_wmma_f32_16x16x4_f32(false, a, false, br, (short)0, accR, false, false);
        accI = __builtin_amdgcn_wmma_f32_16x16x4_f32(false, a, false, bi, (short)0, accI, false, false);
      }
#pragma unroll
      for (int r = 0; r < 8; ++r) {    // C/D layout: m = r + 8*hi, n = l15
        const int m = m0 + r + hi * 8;
        buLds[m * N_ + n0 + l15] = make_float2(accR[r], accI[r]);
      }
    }
    __syncthreads();

    // ---- chunk-local scan: thread = channel, sequential over SUB steps ----
    float2* dst = states + ((size_t)b * (L_ + 1) + tbase + s * SUB + 1) * N_ + tid;
    for (int t = 0; t < SUB; ++t) {
      const float2 v = buLds[t * N_ + tid];
      float2 nx;
      nx.x = lam.x * x.x - lam.y * x.y + v.x;
      nx.y = lam.x * x.y + lam.y * x.x + v.y;
      x = nx;
      dst[(size_t)t * N_] = x;
    }
  }
}

// ---------------------------------------------------------------------------
// k2: exclusive cross-chunk prefix of carries: P_c = carry_{c-1} + lam^T*P_{c-1}
// Grid: B_ blocks x 256 threads (thread = channel). Also zeroes states[b][0].
// ---------------------------------------------------------------------------
__global__ __launch_bounds__(256) void lru_prefix(
    const float2* __restrict__ lampow, const float2* __restrict__ states,
    float2* __restrict__ P, float2* __restrict__ states_w) {
  const int b = blockIdx.x;
  const int n = threadIdx.x;
  const float2 lamT = lampow[(size_t)T_CH * N_ + n];
  float2 p = make_float2(0.f, 0.f);
  states_w[((size_t)b * (L_ + 1)) * N_ + n] = make_float2(0.f, 0.f);  // x0 = 0
  for (int c = 0; c < NCH; ++c) {
    P[((size_t)b * NCH + c) * N_ + n] = p;
    const float2 carry = states[((size_t)b * (L_ + 1) + (size_t)(c + 1) * T_CH) * N_ + n];
    p = make_float2(lamT.x * p.x - lamT.y * p.y + carry.x,
                    lamT.x * p.y + lamT.y * p.x + carry.y);
  }
}

// ---------------------------------------------------------------------------
// k3a: in-place fixup: x[b][t] = y + lam^{(t%T)+1} * P[b][t/T].  One pass.
// ---------------------------------------------------------------------------
__global__ __launch_bounds__(256) void lru_fixup(
    const float2* __restrict__ lampow, const float2* __restrict__ P,
    float2* __restrict__ states) {
  const size_t idx = (size_t)blockIdx.x * 256 + threadIdx.x;  // over B*L*N
  const int n = (int)(idx & (N_ - 1));
  const size_t r = idx >> 8;
  const int t = (int)(r & (L_ - 1));
  const int b = (int)(r >> 13);
  const int c = t / T_CH;
  const int k = (t % T_CH) + 1;
  const size_t si = ((size_t)b * (L_ + 1) + t + 1) * N_ + n;
  const float2 y  = states[si];
  const float2 p  = P[((size_t)b * NCH + c) * N_ + n];
  const float2 lp = lampow[(size_t)k * N_ + n];
  float2 xo;
  xo.x = y.x + lp.x * p.x - lp.y * p.y;
  xo.y = y.y + lp.x * p.y + lp.y * p.x;
  states[si] = xo;
}

// ---------------------------------------------------------------------------
// k3b: output GEMM: out[b][t] = Re(states[b][t] @ C^T) + u[b][t] @ D^T
// (states index t == x_{t-1}).  States+u tiles staged raw via TDM; re/im GEMMs
// use separate accumulators (f32 WMMA has no A-negate) and subtract at the end.
// Grid: (B*L)/SUB blocks of 256 threads.
// ---------------------------------------------------------------------------
__global__ __launch_bounds__(256) void lru_out(
    const float* __restrict__ u, const float2* __restrict__ states,
    const float* __restrict__ Cre, const float* __restrict__ Cim,
    const float* __restrict__ D, float* __restrict__ out) {
  __shared__ float2 XLds[SUB * N_];   // 128 KB, raw interleaved complex rows
  __shared__ float  uLds[SUB * H_];   // 32 KB
  const int tid = threadIdx.x;
  const size_t r0 = (size_t)blockIdx.x * SUB;  // flattened (b,t) row base
  const int b  = (int)(r0 >> 13);
  const int t0 = (int)(r0 & (L_ - 1));

  // warm L2-resident C/D matrices (C: 2x128 KB, D: 64 KB)
  __builtin_prefetch(Cre + (size_t)tid * 128, 0, 0);
  __builtin_prefetch(Cim + (size_t)tid * 128, 0, 0);
  __builtin_prefetch(D + (size_t)tid * 64, 0, 0);

  const float2* xsrc = states + ((size_t)b * (L_ + 1) + t0) * N_;  // contiguous 128 KB
  const float*  usrc = u + ((size_t)b * L_ + t0) * H_;             // contiguous 32 KB
  const int lane = tid & 31, w = tid >> 5;
#ifdef HAS_TDM
  if (w == 0) {
    tdm_copy_to_lds(XLds, xsrc, SUB * N_, 3u /*8B elems*/);
    tdm_copy_to_lds(uLds, usrc, SUB * H_, 2u /*4B elems*/);
    wait_tensorcnt0();
  }
#else
  for (int i = tid; i < SUB * N_; i += 256) XLds[i] = xsrc[i];
  {
    const float4* usrc4 = (const float4*)usrc;
    float4* uLds4 = (float4*)uLds;
    for (int i = tid; i < SUB * H_ / 4; i += 256) uLds4[i] = usrc4[i];
  }
#endif
  __syncthreads();

  const int hi = (lane >> 4) & 1, l15 = lane & 15;
  const int m0 = (w >> 1) * 16;
  const int obase = (w & 1) * 64;
  const float2* ax = &XLds[(m0 + l15) * N_ + 2 * hi];
  const float*  au = &uLds[(m0 + l15) * H_ + 2 * hi];

  for (int j = 0; j < 4; ++j) {
    const int o0 = obase + j * 16;
    v8f acc1 = {0.f, 0.f, 0.f, 0.f, 0.f, 0.f, 0.f, 0.f};  // Xre*Cre + u*D
    v8f acc2 = {0.f, 0.f, 0.f, 0.f, 0.f, 0.f, 0.f, 0.f};  // Xim*Cim
    const float* bcr = &Cre[(size_t)(o0 + l15) * N_ + 2 * hi];
    const float* bci = &Cim[(size_t)(o0 + l15) * N_ + 2 * hi];
    const float* bd  = &D[(size_t)(o0 + l15) * H_ + 2 * hi];
    for (int k0 = 0; k0 < N_; k0 += 4) {
      const float2 c0 = ax[k0];
      const float2 c1 = ax[k0 + 1];
      v2f are; are[0] = c0.x; are[1] = c1.x;
      v2f aim; aim[0] = c0.y; aim[1] = c1.y;
      v2f bre = *(const v2f*)(bcr + k0);
      v2f bim = *(const v2f*)(bci + k0);
      acc1 = __builtin_amdgcn_wmma_f32_16x16x4_f32(false, are, false, bre, (short)0, acc1, false, false);
      acc2 = __builtin_amdgcn_wmma_f32_16x16x4_f32(false, aim, false, bim, (short)0, acc2, false, false);
    }
    for (int k0 = 0; k0 < H_; k0 += 4) {
      v2f a  = *(const v2f*)(au + k0);
      v2f bv = *(const v2f*)(bd + k0);
      acc1 = __builtin_amdgcn_wmma_f32_16x16x4_f32(false, a, false, bv, (short)0, acc1, false, false);
    }
#pragma unroll
    for (int rr = 0; rr < 8; ++rr) {
      const int m = m0 + rr + hi * 8;
      out[(r0 + m) * (size_t)O_ + o0 + l15] = acc1[rr] - acc2[rr];
    }
  }
}

// ---------------------------------------------------------------------------
extern "C" void kernel_launch(void* const* d_in, const int* in_sizes, int n_in,
                              void* d_out, int out_size, void* d_ws, size_t ws_size,
                              hipStream_t stream) {
  const float* u         = (const float*)d_in[0];
  const float* nu_log    = (const float*)d_in[1];
  const float* theta_log = (const float*)d_in[2];
  const float* gamma_log = (const float*)d_in[3];
  const float* C_re      = (const float*)d_in[6];
  const float* C_im      = (const float*)d_in[7];
  const float* Bp_re     = (const float*)d_in[4];
  const float* Bp_im     = (const float*)d_in[5];
  const float* Dm        = (const float*)d_in[8];

  float*  out    = (float*)d_out;
  float2* states = (float2*)(out + (size_t)B_ * L_ * O_);  // c64 trajectory

  // workspace layout
  char*   ws     = (char*)d_ws;
  float*  Bsc_re = (float*)ws;                                   // 128 KB
  float*  Bsc_im = (float*)(ws + (size_t)N_ * H_ * 4);           // 128 KB
  float2* lampow = (float2*)(ws + (size_t)2 * N_ * H_ * 4);      // (T+1)*N c64
  float2* P      = (float2*)(ws + (size_t)2 * N_ * H_ * 4 +
                             (size_t)(T_CH + 1) * N_ * 8);       // B*NCH*N c64

  lru_prep<<<1, 256, 0, stream>>>(nu_log, theta_log, gamma_log, Bp_re, Bp_im,
                                  Bsc_re, Bsc_im, lampow);
  lru_gemm_scan<<<B_ * NCH, 256, 0, stream>>>(u, Bsc_re, Bsc_im, lampow, states);
  lru_prefix<<<B_, 256, 0, stream>>>(lampow, states, P, states);
  lru_fixup<<<(B_ * L_ * N_) / 256, 256, 0, stream>>>(lampow, P, states);
  lru_out<<<(B_ * L_) / SUB, 256, 0, stream>>>(u, states, C_re, C_im, Dm, out);
}